// BaryGeom_36661840838891
// MI455X (gfx1250) — compile-verified
//
#include <hip/hip_runtime.h>
#include <cmath>

// ---------------------------------------------------------------------------
// SPD geometric-mean step (Karcher barycenter, K_ITERS=1) on gfx1250.
// All dense N×N matmuls run through V_WMMA_F32_16X16X4_F32 (wave32 WMMA),
// with every intermediate matrix resident in LDS.
// ---------------------------------------------------------------------------

#define MAXN          64      // max padded matrix dim (4 LDS bufs = 64 KB)
#define NTHREADS      256     // 8 wave32 waves per workgroup
#define NS_ITERS_G    14      // Newton-Schulz iters for global sqrt/isqrt
#define NS_ITERS_R    9       // Newton-Schulz iters per square root in log
#define NROOTS        4       // inverse-scaling square roots for matrix log
#define LOG_ORDER     12      // Mercator series order
#define EXP_ORDER     8       // Taylor order for exp
#define EXP_SQUARINGS 8       // scaling-and-squaring steps

typedef __attribute__((ext_vector_type(2))) float v2f;
typedef __attribute__((ext_vector_type(8))) float v8f;

// One wave computes one 16x16 tile of D = A @ B (f32, K stepped by 4).
// A/B/D may be LDS or global pointers. Caller handles barriers.
// ISA layouts: A 16x4 -> lane m=lane%16, khalf=lane/16 holds K={kb,kb+1};
//              B 4x16 -> lane n=lane%16, khalf holds rows {kb,kb+1};
//              C/D    -> VGPR r: row = r + 8*khalf, col = lane%16.
__device__ __forceinline__ void wg_matmul(float* __restrict__ D,
                                          const float* __restrict__ A,
                                          const float* __restrict__ B, int np) {
  const int lane  = threadIdx.x & 31;
  const int wave  = threadIdx.x >> 5;
  const int nwav  = blockDim.x >> 5;
  const int tc    = np >> 4;
  const int tiles = tc * tc;
  const int khalf = lane >> 4;
  const int ml    = lane & 15;
  for (int t = wave; t < tiles; t += nwav) {
    const int ti  = t / tc;
    const int tj  = t % tc;
    const int row = ti * 16 + ml;
    const int col = tj * 16 + ml;
    v8f c = {};
    for (int k = 0; k < np; k += 4) {
      const int kb = k + 2 * khalf;
      v2f a, b;
      a.x = A[row * np + kb];
      a.y = A[row * np + kb + 1];
      b.x = B[kb * np + col];
      b.y = B[(kb + 1) * np + col];
      // (neg_a, A, neg_b, B, c_mod, C, reuse_a, reuse_b)
      c = __builtin_amdgcn_wmma_f32_16x16x4_f32(false, a, false, b,
                                                (short)0, c, false, false);
    }
#pragma unroll
    for (int r = 0; r < 8; ++r)
      D[(ti * 16 + r + 8 * khalf) * np + col] = c[r];
  }
}

// Block-wide trace reduction using a free LDS buffer as scratch.
__device__ __forceinline__ float wg_trace(const float* A, int np, float* scratch) {
  float v = 0.f;
  for (int i = threadIdx.x; i < np; i += blockDim.x) v += A[i * np + i];
  scratch[threadIdx.x] = v;
  __syncthreads();
  for (int s = blockDim.x >> 1; s > 0; s >>= 1) {
    if ((int)threadIdx.x < s) scratch[threadIdx.x] += scratch[threadIdx.x + s];
    __syncthreads();
  }
  float r = scratch[0];
  __syncthreads();
  return r;
}

// Coupled Newton-Schulz: on entry y=A (SPD, eigs in (0,1]), z=I.
// On exit y=A^{1/2}, z=A^{-1/2}. Pointers rotate; 4 distinct buffers required.
__device__ void wg_ns_pair(float*& y, float*& z, float*& t, float*& f,
                           int np, int iters) {
  const int nn = np * np;
  for (int it = 0; it < iters; ++it) {
    __syncthreads();
    wg_matmul(t, z, y, np);                       // t = Z*Y
    __syncthreads();
    for (int idx = threadIdx.x; idx < nn; idx += blockDim.x) {
      const float d = ((idx / np) == (idx % np)) ? 3.0f : 0.0f;
      t[idx] = 0.5f * (d - t[idx]);               // t = (3I - ZY)/2
    }
    __syncthreads();
    wg_matmul(f, y, t, np);                       // newY = Y*t
    __syncthreads();
    wg_matmul(y, t, z, np);                       // newZ = t*Z (into old-Y buf)
    __syncthreads();
    float* oy = y; float* oz = z;
    y = f; z = oy; f = oz;                        // rotate buffers
  }
}

// --- Kernel 1: G = mean_b X_b, identity-padded to np×np -------------------
__global__ void mean_kernel(const float* __restrict__ x, float* __restrict__ G,
                            int n, int np, int batch, long long stride) {
  const int idx = blockIdx.x * blockDim.x + threadIdx.x;
  const int nn  = np * np;
  if (idx >= nn) return;
  const int i = idx / np, j = idx % np;
  float v;
  if (i < n && j < n) {
    float s = 0.f;
    const long long off = (long long)i * n + j;
    for (int b = 0; b < batch; ++b) s += x[(long long)b * stride + off];
    v = s / (float)batch;
  } else {
    v = (i == j) ? 1.f : 0.f;
  }
  G[idx] = v;
}

// --- Kernel 2: sq = G^{1/2}, isq = G^{-1/2}; zero Mbar --------------------
__global__ void sqrt_kernel(const float* __restrict__ G, float* __restrict__ sq,
                            float* __restrict__ isq, float* __restrict__ Mbar,
                            int np) {
  __shared__ float S0[MAXN * MAXN], S1[MAXN * MAXN], S2[MAXN * MAXN], S3[MAXN * MAXN];
  float* y = S0; float* z = S1; float* t = S2; float* f = S3;
  const int nn = np * np;
  for (int idx = threadIdx.x; idx < nn; idx += blockDim.x) y[idx] = G[idx];
  __syncthreads();
  const float c   = wg_trace(y, np, t);           // eigs(G/c) in (0,1]
  const float inv = 1.f / c;
  for (int idx = threadIdx.x; idx < nn; idx += blockDim.x) {
    y[idx] *= inv;
    z[idx]  = ((idx / np) == (idx % np)) ? 1.f : 0.f;
  }
  __syncthreads();
  wg_ns_pair(y, z, t, f, np, NS_ITERS_G);
  __syncthreads();
  const float sc = sqrtf(c);
  const float rs = 1.f / sc;
  for (int idx = threadIdx.x; idx < nn; idx += blockDim.x) {
    sq[idx]   = sc * y[idx];
    isq[idx]  = rs * z[idx];
    Mbar[idx] = 0.f;
  }
}

// --- Kernel 3 (per batch element): L_b = log(isq·X_b·isq), Mbar += L_b/B --
__global__ void batch_log_kernel(const float* __restrict__ x,
                                 const float* __restrict__ isq,
                                 float* __restrict__ Mbar,
                                 int n, int np, long long stride, float invB) {
  __shared__ float S0[MAXN * MAXN], S1[MAXN * MAXN], S2[MAXN * MAXN], S3[MAXN * MAXN];
  float* y = S0; float* z = S1; float* t = S2; float* f = S3;
  const int nn = np * np;
  const long long base = (long long)blockIdx.x * stride;

  // Load identity-padded X_b.
  for (int idx = threadIdx.x; idx < nn; idx += blockDim.x) {
    const int i = idx / np, j = idx % np;
    y[idx] = (i < n && j < n) ? x[base + (long long)i * n + j]
                              : ((i == j) ? 1.f : 0.f);
  }
  __syncthreads();
  wg_matmul(t, isq, y, np);                       // t = isq * X
  __syncthreads();
  wg_matmul(z, t, isq, np);                       // z = T = isq * X * isq
  __syncthreads();

  const float c    = wg_trace(z, np, f);          // eigs(T/c) in (0,1]
  const float logc = logf(c);
  const float inv  = 1.f / c;
  for (int idx = threadIdx.x; idx < nn; idx += blockDim.x) y[idx] = z[idx] * inv;

  // Inverse scaling: y <- A^{1/2^NROOTS} via repeated Newton-Schulz.
  for (int r = 0; r < NROOTS; ++r) {
    __syncthreads();
    for (int idx = threadIdx.x; idx < nn; idx += blockDim.x)
      z[idx] = ((idx / np) == (idx % np)) ? 1.f : 0.f;
    __syncthreads();
    wg_ns_pair(y, z, t, f, np, NS_ITERS_R);
  }
  __syncthreads();

  // E = y - I; Mercator series  log(I+E) = sum (-1)^{k+1} E^k / k.
  for (int idx = threadIdx.x; idx < nn; idx += blockDim.x)
    if ((idx / np) == (idx % np)) y[idx] -= 1.f;
  __syncthreads();
  for (int idx = threadIdx.x; idx < nn; idx += blockDim.x) {
    t[idx] = y[idx];                              // acc
    f[idx] = y[idx];                              // current power p = E
  }
  for (int k = 2; k <= LOG_ORDER; ++k) {
    __syncthreads();
    wg_matmul(z, f, y, np);                       // z = p * E
    __syncthreads();
    float* tmp = f; f = z; z = tmp;               // p = new power
    const float coef = ((k & 1) ? 1.f : -1.f) / (float)k;
    for (int idx = threadIdx.x; idx < nn; idx += blockDim.x)
      t[idx] += coef * f[idx];
  }
  __syncthreads();

  // L = 2^NROOTS * series + log(c) * I ; accumulate mean.
  const float sc = (float)(1 << NROOTS);
  for (int idx = threadIdx.x; idx < nn; idx += blockDim.x) {
    const float l = sc * t[idx] + (((idx / np) == (idx % np)) ? logc : 0.f);
    atomicAdd(&Mbar[idx], invB * l);
  }
}

// --- Kernel 4: out = sq · exp(Mbar) · sq ----------------------------------
__global__ void exp_final_kernel(const float* __restrict__ Mbar,
                                 const float* __restrict__ sq,
                                 float* __restrict__ out, int n, int np) {
  __shared__ float S0[MAXN * MAXN], S1[MAXN * MAXN], S2[MAXN * MAXN], S3[MAXN * MAXN];
  float* m = S0; float* acc = S1; float* p = S2; float* f = S3;
  const int nn = np * np;
  const float ds = 1.f / (float)(1 << EXP_SQUARINGS);

  for (int idx = threadIdx.x; idx < nn; idx += blockDim.x) m[idx] = Mbar[idx] * ds;
  __syncthreads();
  for (int idx = threadIdx.x; idx < nn; idx += blockDim.x) {
    const float v = m[idx];
    acc[idx] = v + (((idx / np) == (idx % np)) ? 1.f : 0.f);  // I + M
    p[idx]   = v;                                             // P_1 = M
  }
  // Taylor: P_k = P_{k-1} * M / k ; acc += P_k.
  for (int k = 2; k <= EXP_ORDER; ++k) {
    __syncthreads();
    wg_matmul(f, p, m, np);
    __syncthreads();
    const float rk = 1.f / (float)k;
    for (int idx = threadIdx.x; idx < nn; idx += blockDim.x) {
      f[idx] *= rk;
      acc[idx] += f[idx];
    }
    float* tmp = p; p = f; f = tmp;
  }
  // Squaring: acc <- acc^2, EXP_SQUARINGS times.
  for (int s = 0; s < EXP_SQUARINGS; ++s) {
    __syncthreads();
    wg_matmul(p, acc, acc, np);
    __syncthreads();
    float* tmp = acc; acc = p; p = tmp;
  }
  // G = sq * acc * sq.
  wg_matmul(m, sq, acc, np);
  __syncthreads();
  wg_matmul(f, m, sq, np);
  __syncthreads();
  for (int idx = threadIdx.x; idx < n * n; idx += blockDim.x)
    out[idx] = f[(idx / n) * np + (idx % n)];
}

extern "C" void kernel_launch(void* const* d_in, const int* in_sizes, int n_in,
                              void* d_out, int out_size, void* d_ws, size_t ws_size,
                              hipStream_t stream) {
  const float* x = (const float*)d_in[0];
  float* out = (float*)d_out;

  const int n  = (int)(sqrt((double)out_size) + 0.5);
  if (n <= 0) return;
  const int np = (n + 15) & ~15;
  if (np > MAXN) return;  // unsupported size for single-WGP LDS pipeline

  // Assumption: x has shape (B, 1, N, N) so x[:,0] is the full tensor.
  const long long stride = (long long)n * n;
  int batch = (int)(in_sizes[0] / stride);
  if (batch < 1) batch = 1;

  const size_t mat = (size_t)np * np;
  float* ws   = (float*)d_ws;
  float* G    = ws;
  float* sq   = ws + mat;
  float* isq  = ws + 2 * mat;
  float* Mbar = ws + 3 * mat;

  const int nn = np * np;
  mean_kernel<<<(nn + NTHREADS - 1) / NTHREADS, NTHREADS, 0, stream>>>(
      x, G, n, np, batch, stride);
  sqrt_kernel<<<1, NTHREADS, 0, stream>>>(G, sq, isq, Mbar, np);
  batch_log_kernel<<<batch, NTHREADS, 0, stream>>>(
      x, isq, Mbar, n, np, stride, 1.0f / (float)batch);
  exp_final_kernel<<<1, NTHREADS, 0, stream>>>(Mbar, sq, out, n, np);
}